// CorrelationBlock_18726057410753
// MI455X (gfx1250) — compile-verified
//
#include <hip/hip_runtime.h>
#include <hip/hip_bf16.h>

typedef float v2f __attribute__((ext_vector_type(2)));
typedef float v8f __attribute__((ext_vector_type(8)));

#define NPTS      6144      // N = H*W at level 0
#define CDIM      128
#define NCHUNK    512       // points per corr chunk (chunk corr <= 12.6 MB, L2-resident)
#define INV_SQRTC 0.088388347648318447f
#define KWIN      9
#define OUT_PER_PT (4 * KWIN * KWIN)   // 324

// ---------------------------------------------------------------------------
// 2x2 average pool: src [S*C, Hs, Ws] -> dst [S*C, Hs/2, Ws/2]
// ---------------------------------------------------------------------------
__global__ __launch_bounds__(256)
void corrblk_pool2(const float* __restrict__ src, float* __restrict__ dst,
                   int Hs, int Ws, int count) {
    int idx = blockIdx.x * 256 + threadIdx.x;
    if (idx >= count) return;
    int Wd = Ws >> 1, Hd = Hs >> 1;
    int plane = Hd * Wd;
    int sc = idx / plane;
    int r  = idx - sc * plane;
    int y  = r / Wd;
    int x  = r - y * Wd;
    const float* sp = src + (size_t)sc * Hs * Ws + (size_t)(2 * y) * Ws + 2 * x;
    dst[idx] = 0.25f * (sp[0] + sp[1] + sp[Ws] + sp[Ws + 1]);
}

// ---------------------------------------------------------------------------
// Correlation GEMM via fp32 WMMA:  corr[nl, hw] = (1/sqrt(C)) * sum_c A[n0+nl,c]*B[c,hw]
// A: targets slice [N, C] row-major (C contiguous)
// B: feats slice   [C, HW] row-major (HW contiguous)
// One wave computes a 16(M) x 32(N) strip: 2 accumulators, 32 k-steps of K=4.
// ---------------------------------------------------------------------------
__global__ __launch_bounds__(256)
void corrblk_gemm(const float* __restrict__ A, const float* __restrict__ B,
                  float* __restrict__ corr, int HW, int n0, int totalTiles) {
    int wave = blockIdx.x * (256 / 32) + (threadIdx.x >> 5);
    if (wave >= totalTiles) return;
    int lane = threadIdx.x & 31;

    int ntiles = HW >> 5;               // 32-wide N tiles
    int mt = wave / ntiles;             // 16-row M tile within chunk
    int nt = wave - mt * ntiles;

    int m   = lane & 15;
    int hi  = lane >> 4;                // 0: K pair {0,1}, 1: K pair {2,3}
    int col = nt * 32 + (lane & 15);

    const float* arow = A + (size_t)(n0 + mt * 16 + m) * CDIM;

    v8f acc0 = {0.f, 0.f, 0.f, 0.f, 0.f, 0.f, 0.f, 0.f};
    v8f acc1 = {0.f, 0.f, 0.f, 0.f, 0.f, 0.f, 0.f, 0.f};

#pragma unroll 4
    for (int kk = 0; kk < CDIM / 4; ++kk) {
        int kb = kk * 4 + hi * 2;
        // A fragment: this lane holds A[m, kb] and A[m, kb+1]
        v2f a = *(const v2f*)(arow + kb);
        // B fragments: this lane holds B[kb, col] and B[kb+1, col]
        const float* bp = B + (size_t)kb * HW + col;
        v2f b0, b1;
        b0.x = bp[0];       b0.y = bp[HW];
        b1.x = bp[16];      b1.y = bp[HW + 16];
        acc0 = __builtin_amdgcn_wmma_f32_16x16x4_f32(false, a, false, b0,
                                                     (short)0, acc0, false, false);
        acc1 = __builtin_amdgcn_wmma_f32_16x16x4_f32(false, a, false, b1,
                                                     (short)0, acc1, false, false);
    }

    // D layout: VGPR r -> row (hi ? 8+r : r), column = lane&15 within tile
    int rowBase = mt * 16 + hi * 8;
#pragma unroll
    for (int r = 0; r < 8; ++r) {
        corr[(size_t)(rowBase + r) * HW + col]      = acc0[r] * INV_SQRTC;
        corr[(size_t)(rowBase + r) * HW + col + 16] = acc1[r] * INV_SQRTC;
    }
}

// ---------------------------------------------------------------------------
// Bilinear window sampling from the L2-resident corr chunk.
// One thread per (point-in-chunk, window element). Zero padding.
// Reference convention: x = cx/2^l + d[ki], y = cy/2^l + d[kj],
// output slot = level*81 + ki*9 + kj.
// ---------------------------------------------------------------------------
__global__ __launch_bounds__(256)
void corrblk_sample(const float* __restrict__ corr, const float* __restrict__ coords,
                    float* __restrict__ out, int n0, int Hc, int Wc,
                    int level, int s, float invScale) {
    int idx = blockIdx.x * 256 + threadIdx.x;
    if (idx >= NCHUNK * KWIN * KWIN) return;

    int nl = idx / (KWIN * KWIN);
    int e  = idx - nl * (KWIN * KWIN);
    int ki = e / KWIN;
    int kj = e - ki * KWIN;

    int n = n0 + nl;
    float cx = coords[((size_t)(s * NPTS + n)) * 2 + 0];
    float cy = coords[((size_t)(s * NPTS + n)) * 2 + 1];

    float fx = cx * invScale + (float)(ki - 4);
    float fy = cy * invScale + (float)(kj - 4);

    float x0f = floorf(fx), y0f = floorf(fy);
    int x0 = (int)x0f, y0 = (int)y0f;
    int x1 = x0 + 1,   y1 = y0 + 1;
    float wx1 = fx - x0f, wx0 = 1.f - wx1;
    float wy1 = fy - y0f, wy0 = 1.f - wy1;

    const float* crow = corr + (size_t)nl * Hc * Wc;
    float v = 0.f;
    if (x0 >= 0 && x0 < Wc && y0 >= 0 && y0 < Hc) v += crow[y0 * Wc + x0] * wx0 * wy0;
    if (x1 >= 0 && x1 < Wc && y0 >= 0 && y0 < Hc) v += crow[y0 * Wc + x1] * wx1 * wy0;
    if (x0 >= 0 && x0 < Wc && y1 >= 0 && y1 < Hc) v += crow[y1 * Wc + x0] * wx0 * wy1;
    if (x1 >= 0 && x1 < Wc && y1 >= 0 && y1 < Hc) v += crow[y1 * Wc + x1] * wx1 * wy1;

    out[(size_t)(s * NPTS + n) * OUT_PER_PT + level * (KWIN * KWIN) + e] = v;
}

// ---------------------------------------------------------------------------
// Host-side orchestration (graph-capture safe: only kernel launches on stream)
// ---------------------------------------------------------------------------
extern "C" void kernel_launch(void* const* d_in, const int* in_sizes, int n_in,
                              void* d_out, int out_size, void* d_ws, size_t ws_size,
                              hipStream_t stream) {
    const float* features = (const float*)d_in[0]; // [1,2,128,64,96]
    const float* targets  = (const float*)d_in[1]; // [1,2,6144,128]
    const float* coords   = (const float*)d_in[2]; // [1,2,6144,2]
    float* out = (float*)d_out;
    float* ws  = (float*)d_ws;

    const int S = 2;
    const int Hl[4] = {64, 32, 16, 8};
    const int Wl[4] = {96, 48, 24, 12};

    // workspace layout: pyramid levels 1..3 (516096 floats), then corr chunk
    float* pyr = ws;
    float* corrChunk = ws + 524288; // 2 MB offset, 8-byte aligned region

    // ---- build pyramid (levels 1..3) ----
    const float* lvlBase[4];
    lvlBase[0] = features;
    {
        const float* src = features;
        size_t off = 0;
        for (int l = 1; l < 4; ++l) {
            int Hs = Hl[l - 1], Ws_ = Wl[l - 1];
            int count = S * CDIM * Hl[l] * Wl[l];
            corrblk_pool2<<<(count + 255) / 256, 256, 0, stream>>>(src, pyr + off, Hs, Ws_, count);
            lvlBase[l] = pyr + off;
            src = pyr + off;
            off += (size_t)count;
        }
    }

    // ---- per (s, level, chunk): WMMA GEMM into ws, then bilinear sampling ----
    for (int s = 0; s < S; ++s) {
        const float* A = targets + (size_t)s * NPTS * CDIM;
        for (int l = 0; l < 4; ++l) {
            int HW = Hl[l] * Wl[l];
            const float* B = lvlBase[l] + (size_t)s * CDIM * HW;
            float invScale = 1.0f / (float)(1 << l);
            for (int n0 = 0; n0 < NPTS; n0 += NCHUNK) {
                int totalTiles = (NCHUNK / 16) * (HW / 32);     // waves needed
                int blocks = (totalTiles + 7) / 8;              // 8 waves per 256-thread block
                corrblk_gemm<<<blocks, 256, 0, stream>>>(A, B, corrChunk, HW, n0, totalTiles);

                int sthreads = NCHUNK * KWIN * KWIN;            // 41472 = 162*256
                corrblk_sample<<<(sthreads + 255) / 256, 256, 0, stream>>>(
                    corrChunk, coords, out, n0, Hl[l], Wl[l], l, s, invScale);
            }
        }
    }
}